// NonIntersect_68487548502782
// MI455X (gfx1250) — compile-verified
//
#include <hip/hip_runtime.h>
#include <hip/hip_bf16.h>

typedef float v2f __attribute__((ext_vector_type(2)));
typedef float v8f __attribute__((ext_vector_type(8)));

#define W_EXP   5.0f
#define GAMMA   0.02f
#define NBATCH  16
#define N1      2048
#define N2      4096
#define ROWT    (N1 / 16)   // 128 row tiles per batch (one per wave)
#define COLT    (N2 / 16)   // 256 column tiles per wave
#define WPB     4           // waves per block (128 threads, wave32)

// One wave = one 16-row query tile of one batch. Sweeps all 256 column tiles
// of xyz2 with V_WMMA_F32_16X16X4_F32 computing S = d^2 - |x1|^2 via the
// [x,y,z,1] x [-2x2,-2y2,-2z2,|x2|^2] embedding; argmin over columns is
// invariant to the dropped per-row constant.
__global__ __launch_bounds__(128) void nn_pen_kernel(
    const float* __restrict__ xyz1,
    const float* __restrict__ xyz2,
    const float* __restrict__ nxyz2,
    float* __restrict__ partial)
{
    __shared__ float    s_val[WPB][16][16];
    __shared__ unsigned s_idx[WPB][16][16];

    const int lane = threadIdx.x & 31;
    const int wv   = threadIdx.x >> 5;
    const int gw   = blockIdx.x * WPB + wv;     // global wave id: 0..2047
    const int b    = gw / ROWT;
    const int rt   = gw % ROWT;
    const int row_base = rt * 16;
    const int l16  = lane & 15;
    const bool hi  = lane >= 16;

    const float* __restrict__ x1b = xyz1  + (size_t)b * N1 * 3;
    const float* __restrict__ x2b = xyz2  + (size_t)b * N2 * 3;
    const float* __restrict__ n2b = nxyz2 + (size_t)b * N2 * 3;

    // ---- A matrix (16x4 f32): rows = query points, K = [x, y, z, 1] ----
    // Lanes 0-15: VGPR0=K0(x), VGPR1=K1(y). Lanes 16-31: VGPR0=K2(z), VGPR1=K3(1).
    const float* pa = x1b + (size_t)(row_base + l16) * 3;
    const float ax = pa[0], ay = pa[1], az = pa[2];
    v2f amat;
    amat.x = hi ? az   : ax;
    amat.y = hi ? 1.0f : ay;

    float    minv[8];
    unsigned mini[8];
#pragma unroll
    for (int r = 0; r < 8; ++r) { minv[r] = __builtin_inff(); mini[r] = 0u; }

    for (int t = 0; t < COLT; ++t) {
        const int j = t * 16 + l16;
        const float* pb = x2b + (size_t)j * 3;
        // prefetch next tile's same-lane point (speculative; harmless at the tail)
        __builtin_prefetch(pb + 48, 0, 3);
        const float px = pb[0], py = pb[1], pz = pb[2];

        // ---- B matrix (4x16 f32): K rows = [-2x2, -2y2, -2z2, |x2|^2] ----
        v2f bmat;
        bmat.x = hi ? (-2.0f * pz) : (-2.0f * px);
        bmat.y = hi ? (px * px + py * py + pz * pz) : (-2.0f * py);

        v8f c = {};
        v8f d = __builtin_amdgcn_wmma_f32_16x16x4_f32(
            /*neg_a=*/false, amat, /*neg_b=*/false, bmat,
            /*c_mod=*/(short)0, c, /*reuse_a=*/false, /*reuse_b=*/false);

        // Column index of every element this lane holds is t*16 + (lane&15).
        const unsigned cand = (unsigned)(t * 16 + l16);
#pragma unroll
        for (int r = 0; r < 8; ++r) {
            const float s = d[r];
            if (s < minv[r]) { minv[r] = s; mini[r] = cand; }  // strict < => first min per lane
        }
    }

    // ---- Stage (min, idx) as a 16x16 matrix in LDS for per-row reduction ----
#pragma unroll
    for (int r = 0; r < 8; ++r) {
        const int m = r + (hi ? 8 : 0);   // C/D layout: slot r = row r (lo half) / r+8 (hi half)
        s_val[wv][m][l16] = minv[r];
        s_idx[wv][m][l16] = mini[r];
    }
    __syncthreads();

    float e = 0.0f;
    if (lane < 16) {
        const int m = lane;               // this lane reduces row m
        float    best = s_val[wv][m][0];
        unsigned bidx = s_idx[wv][m][0];
#pragma unroll
        for (int n = 1; n < 16; ++n) {
            const float    v  = s_val[wv][m][n];
            const unsigned ii = s_idx[wv][m][n];
            if (v < best || (v == best && ii < bidx)) { best = v; bidx = ii; }
        }
        // Epilogue: signed penetration depth against NN's surface element.
        const float* q  = x2b + (size_t)bidx * 3;
        const float* nn = n2b + (size_t)bidx * 3;
        const float* p1 = x1b + (size_t)(row_base + m) * 3;
        const float dps = nn[0] * (q[0] - p1[0])
                        + nn[1] * (q[1] - p1[1])
                        + nn[2] * (q[2] - p1[2]);
        const float tp = fmaxf(dps, 0.0f);
        e = __expf(W_EXP * tp);
    }

    // Wave-level sum (upper half contributes 0), deterministic per-wave partial.
#pragma unroll
    for (int off = 16; off >= 1; off >>= 1)
        e += __shfl_xor(e, off, 32);
    if (lane == 0) partial[gw] = e;
}

// One block per batch: reduce the 128 per-wave partials deterministically.
__global__ __launch_bounds__(128) void finalize_kernel(
    const float* __restrict__ partial, float* __restrict__ out)
{
    __shared__ float s[WPB];
    const int b    = blockIdx.x;
    const int lane = threadIdx.x & 31;
    const int wv   = threadIdx.x >> 5;

    float v = partial[b * ROWT + threadIdx.x];
#pragma unroll
    for (int off = 16; off >= 1; off >>= 1)
        v += __shfl_xor(v, off, 32);
    if (lane == 0) s[wv] = v;
    __syncthreads();
    if (threadIdx.x == 0) {
        const float tot = s[0] + s[1] + s[2] + s[3];
        out[b] = GAMMA * tot / (float)N1;
    }
}

extern "C" void kernel_launch(void* const* d_in, const int* in_sizes, int n_in,
                              void* d_out, int out_size, void* d_ws, size_t ws_size,
                              hipStream_t stream) {
    (void)in_sizes; (void)n_in; (void)out_size; (void)ws_size;
    const float* xyz1  = (const float*)d_in[0];
    const float* xyz2  = (const float*)d_in[1];
    const float* nxyz2 = (const float*)d_in[2];
    float* out     = (float*)d_out;
    float* partial = (float*)d_ws;            // NBATCH*ROWT = 2048 floats

    const int total_waves = NBATCH * ROWT;    // 2048
    nn_pen_kernel<<<total_waves / WPB, WPB * 32, 0, stream>>>(xyz1, xyz2, nxyz2, partial);
    finalize_kernel<<<NBATCH, ROWT, 0, stream>>>(partial, out);
}